// ModelNew_63806034150029
// MI455X (gfx1250) — compile-verified
//
#include <hip/hip_runtime.h>

typedef float v2f __attribute__((ext_vector_type(2)));
typedef float v8f __attribute__((ext_vector_type(8)));

#define CIN    32
#define COUT   32
#define HIN    256
#define WIN    256
#define HOUT   254
#define WOUT   254
#define NIMG   16
#define NEG_SLOPE 0.01f

// Implicit-GEMM 3x3 VALID conv using fp32 WMMA (V_WMMA_F32_16X16X4_F32).
//   GEMM view:  D[cout][pixel] += A[cout][k] * B[k][pixel],  k = (kh,kw,ci), K=288
//   Per wave: one 16-pixel strip (along W) x all 32 couts (two 16x16 accumulators).
__global__ __launch_bounds__(256) void conv3x3_wmma_f32(
    const float* __restrict__ x, const float* __restrict__ wgt,
    const float* __restrict__ bias, float* __restrict__ out)
{
    // Weights re-laid-out for WMMA-A fragments: wlds[kh*3+kw][co][ci]
    __shared__ float wlds[9 * COUT * CIN];   // 36 KB
    __shared__ float blds[COUT];

    const int tid = threadIdx.x;

    // Cooperative stage + reorder OIHW -> [khkw][co][ci]
    for (int i = tid; i < 9 * COUT * CIN; i += 256) {
        int co   = i / (CIN * 9);
        int r    = i - co * (CIN * 9);
        int ci   = r / 9;
        int khkw = r - ci * 9;
        wlds[(khkw * COUT + co) * CIN + ci] = wgt[i];
    }
    if (tid < COUT) blds[tid] = bias[tid];
    __syncthreads();

    const int wave = tid >> 5;
    const int lane = tid & 31;
    const int lh   = lane >> 4;   // half-wave: selects K pair (j=0,1 vs j=2,3) and M+8 in C/D
    const int ll   = lane & 15;   // pixel column within strip / matrix row within fragment

    // strip id -> (img, h, w0)
    const int strip  = blockIdx.x * 8 + wave;        // 16*254*16 strips total
    const int sInRow = strip & 15;                   // 16 strips per output row
    const int row    = strip >> 4;
    const int h      = row % HOUT;
    const int img    = row / HOUT;
    const int w0     = sInRow * 16;

    v8f c0 = {};   // couts 0..15  x pixels w0..w0+15
    v8f c1 = {};   // couts 16..31 x pixels w0..w0+15

    const float* xin = x + (size_t)(img * CIN) * (HIN * WIN);

    for (int khkw = 0; khkw < 9; ++khkw) {
        const int kh = khkw / 3;
        const int kw = khkw - kh * 3;
        int col = w0 + ll + kw;
        col = (col > WIN - 1) ? (WIN - 1) : col;     // clamp tail-strip reads in-bounds
        const float* xrow = xin + (h + kh) * WIN + col;
        const float* wrow = &wlds[khkw * COUT * CIN];

        #pragma unroll
        for (int cs = 0; cs < 8; ++cs) {
            const int ci = cs * 4 + 2 * lh;          // this half-wave's K pair (ci, ci+1)

            // A fragments (16x4 fp32): lane row M=ll, K pair per half-wave.
            // Aligned 8B LDS loads (ci even).
            v2f a0 = *(const v2f*)&wrow[(ll)      * CIN + ci];
            v2f a1 = *(const v2f*)&wrow[(16 + ll) * CIN + ci];

            // B fragment (4x16 fp32): lane column N=ll, K pair per half-wave.
            // Coalesced across lanes (contiguous along W).
            v2f b;
            b.x = xrow[(ci)     * (HIN * WIN)];
            b.y = xrow[(ci + 1) * (HIN * WIN)];

            c0 = __builtin_amdgcn_wmma_f32_16x16x4_f32(false, a0, false, b,
                                                       (short)0, c0, false, false);
            c1 = __builtin_amdgcn_wmma_f32_16x16x4_f32(false, a1, false, b,
                                                       (short)0, c1, false, false);
        }
    }

    // Fused epilogue: bias, /2, LeakyReLU; coalesced stores (N=pixel across lanes).
    const int  wcol  = w0 + ll;
    const bool valid = wcol < WOUT;
    float* obase = out + (size_t)((img * COUT) * HOUT + h) * WOUT + wcol;

    #pragma unroll
    for (int v = 0; v < 8; ++v) {
        const int co0 = v + 8 * lh;       // D layout: lanes 0-15 -> M=v, lanes 16-31 -> M=v+8
        const int co1 = co0 + 16;
        float y0 = (c0[v] + blds[co0]) * 0.5f;
        float y1 = (c1[v] + blds[co1]) * 0.5f;
        y0 = (y0 >= 0.0f) ? y0 : y0 * NEG_SLOPE;
        y1 = (y1 >= 0.0f) ? y1 : y1 * NEG_SLOPE;
        if (valid) {
            obase[(size_t)co0 * (HOUT * WOUT)] = y0;
            obase[(size_t)co1 * (HOUT * WOUT)] = y1;
        }
    }
}

extern "C" void kernel_launch(void* const* d_in, const int* in_sizes, int n_in,
                              void* d_out, int out_size, void* d_ws, size_t ws_size,
                              hipStream_t stream) {
    const float* x  = (const float*)d_in[0];
    const float* w  = (const float*)d_in[1];
    const float* b  = (const float*)d_in[2];
    float* out      = (float*)d_out;

    const int strips = NIMG * HOUT * 16;   // 65024, divisible by 8
    dim3 grid(strips / 8);
    dim3 block(256);
    hipLaunchKernelGGL(conv3x3_wmma_f32, grid, block, 0, stream, x, w, b, out);
}